// MultiDirectionalSSM_47115791237289
// MI455X (gfx1250) — compile-verified
//
#include <hip/hip_runtime.h>
#include <hip/hip_bf16.h>

#define D_MODEL   128
#define D_STATE   64
#define D_CONV    4
#define D_INNER   256
#define NHEADS    4
#define HEADDIM   64
#define CONV_DIM  384           // D_INNER + 2*D_STATE
#define D_PROJ    644           // 2*D_INNER + 2*D_STATE + NHEADS
#define ZX_LD     656           // D_PROJ padded to multiple of 16
#define N_DIR     6
#define NUM_LAYERS 2
#define FUSE_DIM  768           // N_DIR * D_MODEL

typedef __attribute__((ext_vector_type(2))) float v2f;
typedef __attribute__((ext_vector_type(8))) float v8f;

// ---------------------------------------------------------------------------
// WMMA helper: 16x16 f32 tile, D = A(16xK) * B(K x 16) + C using
// V_WMMA_F32_16X16X4_F32.  A: row-major activations (lda = K stride).
// W: weight rows are output columns, i.e. W[n][k], ldw = K.  All loads are
// unconditional (ragged N handled by zero-padded weights) so the inner loop
// is pure global_load_b64 + v_wmma with full EXEC (ISA 7.12 requirement).
// Fragment layouts per CDNA5 ISA 7.12.2:
//   A (16x4 f32): lane l -> M = l&15, VGPR v -> K = 2*(l>>4) + v
//   B (4x16 f32): lane l -> N = l&15, VGPR v -> K = 2*(l>>4) + v
//   C (16x16):    lane l -> N = l&15, VGPR v -> M = v + 8*(l>>4)
// ---------------------------------------------------------------------------
__device__ __forceinline__ v8f wmma_tile_f32(const float* __restrict__ A, int lda,
                                             const float* __restrict__ W, int ldw,
                                             int K, int lo, int hi, v8f c) {
  const float* ap = A + lo * lda + 2 * hi;
  const float* wp = W + lo * ldw + 2 * hi;
#pragma unroll 4
  for (int k0 = 0; k0 < K; k0 += 4) {
    v2f a, b;
    a.x = ap[k0];
    a.y = ap[k0 + 1];
    b.x = wp[k0];
    b.y = wp[k0 + 1];
    c = __builtin_amdgcn_wmma_f32_16x16x4_f32(false, a, false, b, (short)0, c,
                                              false, false);
  }
  return c;
}

// ---------------------------------------------------------------------------
// Pad in_w (12 x 644 x 128) into wpad (12 x 656 x 128), zero rows >= 644,
// so the in-proj GEMM needs no boundary predication.
// ---------------------------------------------------------------------------
__global__ void pad_inw_kernel(const float* __restrict__ in_w,
                               float* __restrict__ wpad) {
  long gid = (long)blockIdx.x * blockDim.x + threadIdx.x;
  const long total = (long)N_DIR * NUM_LAYERS * ZX_LD * D_MODEL;
  if (gid >= total) return;
  int k  = (int)(gid % D_MODEL);
  int n  = (int)((gid / D_MODEL) % ZX_LD);
  int dj = (int)(gid / ((long)D_MODEL * ZX_LD));
  wpad[gid] = (n < D_PROJ) ? in_w[((long)dj * D_PROJ + n) * D_MODEL + k] : 0.0f;
}

// ---------------------------------------------------------------------------
// Single-workgroup in-LDS bitonic sort of packed coordinate keys.
// 16384 keys * 8B = 128KB -> fits in 320KB CDNA5 LDS.
// key = c[a0]<<34 | c[a1]<<24 | c[a2]<<14 | index  (stable lexsort)
// ---------------------------------------------------------------------------
__global__ __launch_bounds__(1024) void sort_kernel(
    const int* __restrict__ coords, int a0, int a1, int a2,
    int* __restrict__ idx, int L) {
  extern __shared__ unsigned long long sk[];
  for (int i = threadIdx.x; i < L; i += blockDim.x) {
    unsigned long long c0 = (unsigned long long)(unsigned)coords[i * 4 + a0];
    unsigned long long c1 = (unsigned long long)(unsigned)coords[i * 4 + a1];
    unsigned long long c2 = (unsigned long long)(unsigned)coords[i * 4 + a2];
    sk[i] = (c0 << 34) | (c1 << 24) | (c2 << 14) | (unsigned long long)i;
  }
  __syncthreads();
  for (int k = 2; k <= L; k <<= 1) {
    for (int j = k >> 1; j > 0; j >>= 1) {
      for (int i = threadIdx.x; i < L; i += blockDim.x) {
        int ixj = i ^ j;
        if (ixj > i) {
          unsigned long long x = sk[i], y = sk[ixj];
          bool asc = ((i & k) == 0);
          if ((x > y) == asc) { sk[i] = y; sk[ixj] = x; }
        }
      }
      __syncthreads();
    }
  }
  for (int i = threadIdx.x; i < L; i += blockDim.x)
    idx[i] = (int)(sk[i] & 0x3FFFull);
}

// gather: x_sorted[i] = vectors[idx[i]]
__global__ void gather_kernel(const float* __restrict__ vec,
                              const int* __restrict__ idx,
                              float* __restrict__ x, int L) {
  int gid = blockIdx.x * blockDim.x + threadIdx.x;
  if (gid >= L * D_MODEL) return;
  int i = gid >> 7, c = gid & 127;
  x[(long)i * D_MODEL + c] = vec[(long)idx[i] * D_MODEL + c];
}

// scatter into the fused concat buffer: multi[idx[i]][d*128 + c] = x[i][c]
__global__ void scatter_kernel(const float* __restrict__ x,
                               const int* __restrict__ idx,
                               float* __restrict__ multi, int d, int L) {
  int gid = blockIdx.x * blockDim.x + threadIdx.x;
  if (gid >= L * D_MODEL) return;
  int i = gid >> 7, c = gid & 127;
  multi[(long)idx[i] * FUSE_DIM + d * D_MODEL + c] = x[(long)i * D_MODEL + c];
}

// LayerNorm, one row per block (C = 128 or 768)
__global__ __launch_bounds__(128) void ln_kernel(const float* __restrict__ x,
                                                 const float* __restrict__ w,
                                                 const float* __restrict__ b,
                                                 float* __restrict__ out, int C) {
  int row = blockIdx.x;
  const float* xr = x + (long)row * C;
  float* orow = out + (long)row * C;
  __shared__ float red[128];
  float s = 0.f;
  for (int c = threadIdx.x; c < C; c += 128) s += xr[c];
  red[threadIdx.x] = s; __syncthreads();
  for (int st = 64; st > 0; st >>= 1) {
    if (threadIdx.x < st) red[threadIdx.x] += red[threadIdx.x + st];
    __syncthreads();
  }
  float mu = red[0] / (float)C;
  __syncthreads();
  float v = 0.f;
  for (int c = threadIdx.x; c < C; c += 128) { float d = xr[c] - mu; v += d * d; }
  red[threadIdx.x] = v; __syncthreads();
  for (int st = 64; st > 0; st >>= 1) {
    if (threadIdx.x < st) red[threadIdx.x] += red[threadIdx.x + st];
    __syncthreads();
  }
  float rs = rsqrtf(red[0] / (float)C + 1e-5f);
  __syncthreads();
  for (int c = threadIdx.x; c < C; c += 128)
    orow[c] = (xr[c] - mu) * rs * w[c] + b[c];
}

// in_proj: zx[L][ZX_LD] = xn[L][128] @ wpad^T (wpad: [656][128], zero-padded)
__global__ __launch_bounds__(256) void gemm_inproj(const float* __restrict__ xn,
                                                   const float* __restrict__ w,
                                                   float* __restrict__ zx, int L) {
  const int NT = ZX_LD / 16;  // 41 col tiles
  int lane = threadIdx.x & 31, wv = threadIdx.x >> 5;
  int tile = blockIdx.x * 8 + wv;
  if (tile >= (L / 16) * NT) return;
  int rt = tile / NT, ct = tile % NT;
  int lo = lane & 15, hi = lane >> 4;
  int colBase = ct * 16;
  v8f c = {};
  c = wmma_tile_f32(xn + (long)rt * 16 * D_MODEL, D_MODEL,
                    w + (long)colBase * D_MODEL, D_MODEL, D_MODEL,
                    lo, hi, c);
  float* out = zx + (long)(rt * 16) * ZX_LD + colBase;
#pragma unroll
  for (int v = 0; v < 8; ++v)
    out[(long)(v + 8 * hi) * ZX_LD + lo] = c[v];
}

// causal depthwise conv(4) + bias + SiLU on xBC part, softplus(dt) part
__global__ void conv_dt_kernel(const float* __restrict__ zx,
                               const float* __restrict__ cw,
                               const float* __restrict__ cb,
                               const float* __restrict__ dt_bias,
                               float* __restrict__ xBC,
                               float* __restrict__ dtb, int L) {
  long gid = (long)blockIdx.x * blockDim.x + threadIdx.x;
  if (gid >= (long)L * (CONV_DIM + NHEADS)) return;
  int i = (int)(gid / (CONV_DIM + NHEADS));
  int c = (int)(gid % (CONV_DIM + NHEADS));
  if (c < CONV_DIM) {
    float acc = cb[c];
#pragma unroll
    for (int t = 0; t < D_CONV; ++t) {
      int r = i - (D_CONV - 1) + t;
      if (r >= 0) acc += cw[c * D_CONV + t] * zx[(long)r * ZX_LD + D_INNER + c];
    }
    xBC[(long)i * CONV_DIM + c] = acc / (1.f + __expf(-acc));   // SiLU
  } else {
    int h = c - CONV_DIM;
    float v = zx[(long)i * ZX_LD + (D_INNER + CONV_DIM) + h] + dt_bias[h];
    dtb[(long)i * NHEADS + h] = (v > 20.f) ? v : log1pf(__expf(v));  // softplus
  }
}

// Sequential SSM scan: one block per head, 256 lanes; lane holds 16 state
// elements (p = tid>>2, n in [(tid&3)*16, +16)).  B/C/xs/dt staged in LDS,
// C.h contraction finished with wave32 shfl_xor.
__global__ __launch_bounds__(256) void scan_kernel(const float* __restrict__ xBC,
                                                   const float* __restrict__ dtb,
                                                   const float* __restrict__ A_log,
                                                   const float* __restrict__ D_skip,
                                                   float* __restrict__ y, int L) {
  int hd = blockIdx.x;
  int tid = threadIdx.x;
  int p = tid >> 2;
  int nb = (tid & 3) * 16;
  float A = -__expf(A_log[hd]);
  float Dh = D_skip[hd];
  float h[16];
#pragma unroll
  for (int r = 0; r < 16; ++r) h[r] = 0.f;
  __shared__ float sB[D_STATE], sC[D_STATE], sxs[HEADDIM], sdt[1];
  for (int i = 0; i < L; ++i) {
    const float* row = xBC + (long)i * CONV_DIM;
    if (tid < 64)        sB[tid]        = row[D_INNER + tid];
    else if (tid < 128)  sC[tid - 64]   = row[D_INNER + D_STATE + (tid - 64)];
    else if (tid < 192)  sxs[tid - 128] = row[hd * HEADDIM + (tid - 128)];
    else if (tid == 192) sdt[0]         = dtb[(long)i * NHEADS + hd];
    __syncthreads();
    float dt = sdt[0];
    float dA = __expf(dt * A);
    float xs = sxs[p];
    float dtx = dt * xs;
    float acc = 0.f;
#pragma unroll
    for (int r = 0; r < 16; ++r) {
      h[r] = dA * h[r] + dtx * sB[nb + r];
      acc += sC[nb + r] * h[r];
    }
    acc += __shfl_xor(acc, 1, 32);
    acc += __shfl_xor(acc, 2, 32);
    if ((tid & 3) == 0)
      y[(long)i * D_INNER + hd * HEADDIM + p] = acc + Dh * xs;
    __syncthreads();
  }
}

// y = RMSNorm( y * SiLU(z) ) * norm_w   (row of 256, in place)
__global__ __launch_bounds__(256) void gate_rms_kernel(float* __restrict__ y,
                                                       const float* __restrict__ zx,
                                                       const float* __restrict__ nw,
                                                       int L) {
  int row = blockIdx.x, e = threadIdx.x;
  float yv = y[(long)row * D_INNER + e];
  float z = zx[(long)row * ZX_LD + e];
  float g = yv * (z / (1.f + __expf(-z)));
  __shared__ float red[256];
  red[e] = g * g; __syncthreads();
  for (int st = 128; st > 0; st >>= 1) {
    if (e < st) red[e] += red[e + st];
    __syncthreads();
  }
  float scale = rsqrtf(red[0] / (float)D_INNER + 1e-5f);
  y[(long)row * D_INNER + e] = g * scale * nw[e];
}

// out_proj + residual: x += y[L][256] @ out_w^T (out_w: [128][256])
__global__ __launch_bounds__(256) void gemm_outproj(const float* __restrict__ y,
                                                    const float* __restrict__ w,
                                                    float* __restrict__ x, int L) {
  const int NT = D_MODEL / 16;  // 8
  int lane = threadIdx.x & 31, wv = threadIdx.x >> 5;
  int tile = blockIdx.x * 8 + wv;
  if (tile >= (L / 16) * NT) return;
  int rt = tile / NT, ct = tile % NT;
  int lo = lane & 15, hi = lane >> 4;
  int colBase = ct * 16;
  v8f c;
#pragma unroll
  for (int v = 0; v < 8; ++v)  // residual init
    c[v] = x[(long)(rt * 16 + v + 8 * hi) * D_MODEL + colBase + lo];
  c = wmma_tile_f32(y + (long)rt * 16 * D_INNER, D_INNER,
                    w + (long)colBase * D_INNER, D_INNER, D_INNER,
                    lo, hi, c);
#pragma unroll
  for (int v = 0; v < 8; ++v)
    x[(long)(rt * 16 + v + 8 * hi) * D_MODEL + colBase + lo] = c[v];
}

// fuse: out = gelu( hn[L][768] @ fuse_w^T + fuse_b )   (fuse_w: [128][768])
__global__ __launch_bounds__(256) void gemm_fuse(const float* __restrict__ hn,
                                                 const float* __restrict__ w,
                                                 const float* __restrict__ bias,
                                                 float* __restrict__ out, int L) {
  const int NT = D_MODEL / 16;
  int lane = threadIdx.x & 31, wv = threadIdx.x >> 5;
  int tile = blockIdx.x * 8 + wv;
  if (tile >= (L / 16) * NT) return;
  int rt = tile / NT, ct = tile % NT;
  int lo = lane & 15, hi = lane >> 4;
  int colBase = ct * 16;
  v8f c = {};
  c = wmma_tile_f32(hn + (long)rt * 16 * FUSE_DIM, FUSE_DIM,
                    w + (long)colBase * FUSE_DIM, FUSE_DIM, FUSE_DIM,
                    lo, hi, c);
  float b = bias[colBase + lo];
#pragma unroll
  for (int v = 0; v < 8; ++v) {
    float val = c[v] + b;
    val = 0.5f * val * (1.f + erff(val * 0.70710678118654752f));  // exact gelu
    out[(long)(rt * 16 + v + 8 * hi) * D_MODEL + colBase + lo] = val;
  }
}

// ---------------------------------------------------------------------------
extern "C" void kernel_launch(void* const* d_in, const int* in_sizes, int n_in,
                              void* d_out, int out_size, void* d_ws, size_t ws_size,
                              hipStream_t stream) {
  const float* vectors  = (const float*)d_in[0];
  const int*   coords   = (const int*)d_in[1];
  const float* ln_w     = (const float*)d_in[2];
  const float* ln_b     = (const float*)d_in[3];
  const float* in_w     = (const float*)d_in[4];
  const float* conv_w   = (const float*)d_in[5];
  const float* conv_b   = (const float*)d_in[6];
  const float* dt_bias  = (const float*)d_in[7];
  const float* A_log    = (const float*)d_in[8];
  const float* D_skip   = (const float*)d_in[9];
  const float* norm_w   = (const float*)d_in[10];
  const float* out_w    = (const float*)d_in[11];
  const float* f_ln_w   = (const float*)d_in[12];
  const float* f_ln_b   = (const float*)d_in[13];
  const float* fuse_w   = (const float*)d_in[14];
  const float* fuse_b   = (const float*)d_in[15];

  const int L = in_sizes[0] / D_MODEL;  // 16384 (power of two; bitonic sort relies on it)

  // workspace carve-up (reused across directions; stream order serializes)
  char* base = (char*)d_ws;
  size_t off = 0;
  auto carve = [&](size_t bytes) -> char* {
    off = (off + 255) & ~(size_t)255;
    char* p = base + off;
    off += bytes;
    return p;
  };
  int*   idx   = (int*)  carve((size_t)L * sizeof(int));
  float* x     = (float*)carve((size_t)L * D_MODEL  * sizeof(float));
  float* xn    = (float*)carve((size_t)L * D_MODEL  * sizeof(float));
  float* zx    = (float*)carve((size_t)L * ZX_LD    * sizeof(float));
  float* xBC   = (float*)carve((size_t)L * CONV_DIM * sizeof(float));
  float* dtb   = (float*)carve((size_t)L * NHEADS   * sizeof(float));
  float* ybuf  = (float*)carve((size_t)L * D_INNER  * sizeof(float));
  float* multi = (float*)carve((size_t)L * FUSE_DIM * sizeof(float));
  float* hn    = (float*)carve((size_t)L * FUSE_DIM * sizeof(float));
  float* wpad  = (float*)carve((size_t)N_DIR * NUM_LAYERS * ZX_LD * D_MODEL * sizeof(float));
  (void)ws_size; (void)n_in;

  static const int ORD[N_DIR][3] = {{0,1,2},{0,2,1},{1,0,2},{1,2,0},{2,0,1},{2,1,0}};

  const int inTiles  = (L / 16) * (ZX_LD / 16);
  const int outTiles = (L / 16) * (D_MODEL / 16);
  const int inBlocks  = (inTiles + 7) / 8;
  const int outBlocks = (outTiles + 7) / 8;
  const int elemBlocks = (L * D_MODEL + 255) / 256;
  const int convBlocks = (int)(((long)L * (CONV_DIM + NHEADS) + 255) / 256);
  const int padBlocks  = (int)(((long)N_DIR * NUM_LAYERS * ZX_LD * D_MODEL + 255) / 256);

  // one-time weight padding (re-run every call; deterministic)
  pad_inw_kernel<<<padBlocks, 256, 0, stream>>>(in_w, wpad);

  for (int d = 0; d < N_DIR; ++d) {
    sort_kernel<<<1, 1024, (size_t)L * 8, stream>>>(coords, ORD[d][0], ORD[d][1],
                                                    ORD[d][2], idx, L);
    gather_kernel<<<elemBlocks, 256, 0, stream>>>(vectors, idx, x, L);

    for (int j = 0; j < NUM_LAYERS; ++j) {
      const size_t dj = (size_t)(d * NUM_LAYERS + j);
      ln_kernel<<<L, 128, 0, stream>>>(x, ln_w + dj * D_MODEL, ln_b + dj * D_MODEL,
                                       xn, D_MODEL);
      gemm_inproj<<<inBlocks, 256, 0, stream>>>(xn, wpad + dj * ZX_LD * D_MODEL,
                                                zx, L);
      conv_dt_kernel<<<convBlocks, 256, 0, stream>>>(zx, conv_w + dj * CONV_DIM * D_CONV,
                                                     conv_b + dj * CONV_DIM,
                                                     dt_bias + dj * NHEADS, xBC, dtb, L);
      scan_kernel<<<NHEADS, 256, 0, stream>>>(xBC, dtb, A_log + dj * NHEADS,
                                              D_skip + dj * NHEADS, ybuf, L);
      gate_rms_kernel<<<L, 256, 0, stream>>>(ybuf, zx, norm_w + dj * D_INNER, L);
      gemm_outproj<<<outBlocks, 256, 0, stream>>>(ybuf, out_w + dj * D_MODEL * D_INNER,
                                                  x, L);
    }
    scatter_kernel<<<elemBlocks, 256, 0, stream>>>(x, idx, multi, d, L);
  }

  ln_kernel<<<L, 128, 0, stream>>>(multi, f_ln_w, f_ln_b, hn, FUSE_DIM);
  gemm_fuse<<<outBlocks, 256, 0, stream>>>(hn, fuse_w, fuse_b, (float*)d_out, L);
}